// Joiner_30786325578116
// MI455X (gfx1250) — compile-verified
//
#include <hip/hip_runtime.h>

typedef __attribute__((ext_vector_type(2))) float v2f;
typedef __attribute__((ext_vector_type(4))) float v4f;
typedef __attribute__((ext_vector_type(8))) float v8f;

// Problem constants (from reference)
#define BB   8
#define TT   256
#define UU   65
#define DD   640     // K
#define VV   1024    // N
#define M1   (BB*TT) // 2048
#define M2   (BB*UU) // 520

// ---------------------------------------------------------------------------
// Phase 1: f32 WMMA GEMM  C[M,1024] = A[M,640] * Bm[640,1024] (+ bias)
// Block = 128 threads (4 waves). Wave w computes a 64(M) x 16(N) tile at
// nBase = blockIdx.y*64 + w*16, using V_WMMA_F32_16X16X4_F32.
// Rows are clamped (not masked) so EXEC stays all-ones for WMMA; stores are
// bounds-checked.
// ---------------------------------------------------------------------------
__global__ __launch_bounds__(128)
void wmma_gemm_f32(const float* __restrict__ A, const float* __restrict__ Bm,
                   float* __restrict__ C, const float* __restrict__ bias, int M)
{
    const int lane  = threadIdx.x & 31;
    const int wave  = threadIdx.x >> 5;
    const int mTile = blockIdx.x * 64;
    const int nCol  = blockIdx.y * 64 + wave * 16 + (lane & 15);

    const int mLocal = lane & 15;
    const int kOff   = (lane >> 4) << 1;   // 0 for lanes 0-15, 2 for lanes 16-31

    // Clamp A rows so addresses are always in-bounds (WMMA needs full EXEC).
    int r0 = mTile +  0 + mLocal; r0 = (r0 < M) ? r0 : (M - 1);
    int r1 = mTile + 16 + mLocal; r1 = (r1 < M) ? r1 : (M - 1);
    int r2 = mTile + 32 + mLocal; r2 = (r2 < M) ? r2 : (M - 1);
    int r3 = mTile + 48 + mLocal; r3 = (r3 < M) ? r3 : (M - 1);

    const float* a0 = A + (size_t)r0 * DD + kOff;
    const float* a1 = A + (size_t)r1 * DD + kOff;
    const float* a2 = A + (size_t)r2 * DD + kOff;
    const float* a3 = A + (size_t)r3 * DD + kOff;
    const float* bp = Bm + (size_t)kOff * VV + nCol;

    v8f acc0 = {}, acc1 = {}, acc2 = {}, acc3 = {};

    #pragma unroll 2
    for (int k = 0; k < DD; k += 4) {
        v2f fa0 = *(const v2f*)(a0 + k);
        v2f fa1 = *(const v2f*)(a1 + k);
        v2f fa2 = *(const v2f*)(a2 + k);
        v2f fa3 = *(const v2f*)(a3 + k);
        v2f fb;
        fb.x = bp[(size_t)k * VV];
        fb.y = bp[(size_t)k * VV + VV];

        acc0 = __builtin_amdgcn_wmma_f32_16x16x4_f32(false, fa0, false, fb,
                                                     (short)0, acc0, false, false);
        acc1 = __builtin_amdgcn_wmma_f32_16x16x4_f32(false, fa1, false, fb,
                                                     (short)0, acc1, false, false);
        acc2 = __builtin_amdgcn_wmma_f32_16x16x4_f32(false, fa2, false, fb,
                                                     (short)0, acc2, false, false);
        acc3 = __builtin_amdgcn_wmma_f32_16x16x4_f32(false, fa3, false, fb,
                                                     (short)0, acc3, false, false);
    }

    const float bv = bias ? bias[nCol] : 0.0f;
    const int rowHalf = (lane >> 4) * 8;   // C/D layout: M = 8*(lane/16) + r

    #pragma unroll
    for (int r = 0; r < 8; ++r) {
        int row = mTile + rowHalf + r;
        if (row < M)       C[(size_t)row * VV + nCol]        = acc0[r] + bv;
        if (row + 16 < M)  C[(size_t)(row + 16) * VV + nCol] = acc1[r] + bv;
        if (row + 32 < M)  C[(size_t)(row + 32) * VV + nCol] = acc2[r] + bv;
        if (row + 48 < M)  C[(size_t)(row + 48) * VV + nCol] = acc3[r] + bv;
    }
}

// ---------------------------------------------------------------------------
// Phase 2: out[b,t,u,v] = enc_proj[b,t,v] + pred_proj[b,u,v]
// One block per (b,t). Each thread holds a float4 of the enc row in registers
// and loops over u, reading pred rows (L2-resident) and emitting non-temporal
// 128-bit stores (545 MB stream must not thrash L2). Write-bandwidth bound.
// ---------------------------------------------------------------------------
__global__ __launch_bounds__(256)
void joiner_bcast(const float* __restrict__ enc, const float* __restrict__ pred,
                  float* __restrict__ out)
{
    const int t  = blockIdx.x;          // 0..255
    const int b  = blockIdx.y;          // 0..7
    const int bt = b * TT + t;
    const int v  = (threadIdx.x << 2);  // 4 floats per thread

    const v4f e = *(const v4f*)(enc + (size_t)bt * VV + v);
    const float* prow = pred + (size_t)b * UU * VV + v;
    float*       orow = out  + (size_t)bt * UU * VV + v;

    #pragma unroll 4
    for (int u = 0; u < UU; ++u) {
        v4f p = *(const v4f*)(prow + (size_t)u * VV);
        v4f r = e + p;
        __builtin_nontemporal_store(r, (v4f*)(orow + (size_t)u * VV));
    }
}

extern "C" void kernel_launch(void* const* d_in, const int* in_sizes, int n_in,
                              void* d_out, int out_size, void* d_ws, size_t ws_size,
                              hipStream_t stream)
{
    const float* enc  = (const float*)d_in[0];   // (8,256,640)
    const float* pred = (const float*)d_in[1];   // (8,65,640)
    const float* W    = (const float*)d_in[2];   // (1280,1024)
    const float* bias = (const float*)d_in[3];   // (1024,)
    float*       out  = (float*)d_out;           // (8,256,65,1024)

    float* enc_proj  = (float*)d_ws;             // 2048*1024 f32 = 8 MB
    float* pred_proj = enc_proj + (size_t)M1 * VV; // 520*1024 f32 ~ 2 MB

    // GEMM1: enc_proj = encoder_out @ W[:640] + bias
    wmma_gemm_f32<<<dim3(M1 / 64, VV / 64), 128, 0, stream>>>(
        enc, W, enc_proj, bias, M1);

    // GEMM2: pred_proj = predictor_out @ W[640:]
    wmma_gemm_f32<<<dim3((M2 + 63) / 64, VV / 64), 128, 0, stream>>>(
        pred, W + (size_t)DD * VV, pred_proj, nullptr, M2);

    // Broadcast add (bandwidth-bound phase)
    joiner_bcast<<<dim3(TT, BB), 256, 0, stream>>>(enc_proj, pred_proj, out);
}